// GAT_12747462935039
// MI455X (gfx1250) — compile-verified
//
#include <hip/hip_runtime.h>

#define NNODES 50000
#define NEDGES 400000
#define HEADS  4
#define HD     128          // HEADS * 32 per-head dims
#define NEG_SLOPE 0.1f

typedef __attribute__((ext_vector_type(2))) float v2f;
typedef __attribute__((ext_vector_type(8))) float v8f;

// -------- float atomic max via int/uint atomics (total order over floats) ----
__device__ inline void atomicMaxF32(float* addr, float val) {
  if (val >= 0.0f)
    atomicMax(reinterpret_cast<int*>(addr), __float_as_int(val));
  else
    atomicMin(reinterpret_cast<unsigned int*>(addr),
              static_cast<unsigned int>(__float_as_int(val)));
}

// -------- WMMA fp32 GEMM: C[M x 128-ish] = A[M x K] * B[K x N] (+bias) ------
// Compile-time strides: lda = 256, ldb = 128, ldc = 128. One wave32 per
// 16x16 C tile; K unrolled x8 so loads clause up and 8 WMMAs issue per iter.
// A 16x4 f32: lane L holds row M=L%16, K = k0 + 2*(L/16) (+1) in VGPR0/1.
// B 4x16 f32: lane L holds col N=L%16, K = k0 + 2*(L/16) (+1) in VGPR0/1.
// C/D 16x16 f32: VGPR v at lane L -> M = v + 8*(L/16), N = L%16.
template <int K>
__global__ __launch_bounds__(32)
void gemm_wmma_f32(const float* __restrict__ A,
                   const float* __restrict__ B,
                   float* __restrict__ C,
                   const float* __restrict__ bias) {
  const int lane = threadIdx.x & 31;
  const int half = lane >> 4;          // 0: lanes 0-15, 1: lanes 16-31
  const int l16  = lane & 15;
  const int row0 = blockIdx.y * 16;
  const int col0 = blockIdx.x * 16;

  const float* __restrict__ arow = A + (size_t)(row0 + l16) * 256 + half * 2;
  const float* __restrict__ bcol = B + col0 + l16 + (size_t)(half * 2) * 128;

  v8f c = {};
#pragma unroll 8
  for (int k = 0; k < K; k += 4) {
    v2f a = *reinterpret_cast<const v2f*>(arow + k);   // 8B-aligned b64 load
    v2f b;
    b.x = bcol[(size_t)k * 128];
    b.y = bcol[(size_t)k * 128 + 128];
    // 8 args: (neg_a, A, neg_b, B, c_mod, C, reuse_a, reuse_b)
    c = __builtin_amdgcn_wmma_f32_16x16x4_f32(false, a, false, b,
                                              (short)0, c, false, false);
  }

  const float bv = bias ? bias[col0 + l16] : 0.0f;
#pragma unroll
  for (int v = 0; v < 8; ++v) {
    C[(size_t)(row0 + v + half * 8) * 128 + col0 + l16] = c[v] + bv;
  }
}

// -------- elementwise helpers -----------------------------------------------
__global__ void fill4_f32(float4* __restrict__ p, float v, int n4) {
  int i = blockIdx.x * blockDim.x + threadIdx.x;
  if (i < n4) p[i] = make_float4(v, v, v, v);
}

__global__ void init_ms(float* __restrict__ m, float* __restrict__ s, int n) {
  int i = blockIdx.x * blockDim.x + threadIdx.x;
  if (i < n) { m[i] = -1e30f; s[i] = 0.0f; }
}

__global__ void recip_f32(float* __restrict__ s, int n) {
  int i = blockIdx.x * blockDim.x + threadIdx.x;
  if (i < n) s[i] = 1.0f / s[i];   // s==0 only for nodes with no in-edges
}

// el[n,h] = dot(ft[n,h,:], al[h,:]) ; er likewise. One thread per (n,h).
__global__ void attn_logits(const float* __restrict__ ft,
                            const float* __restrict__ al,
                            const float* __restrict__ ar,
                            float* __restrict__ el, float* __restrict__ er) {
  int i = blockIdx.x * blockDim.x + threadIdx.x;   // over NNODES*HEADS
  if (i >= NNODES * HEADS) return;
  int n = i >> 2, h = i & 3;
  const float* f  = ft + (size_t)n * HD + h * 32;
  const float* la = al + h * 32;
  const float* ra = ar + h * 32;
  float sl = 0.0f, sr = 0.0f;
#pragma unroll
  for (int d = 0; d < 32; ++d) {
    float v = f[d];
    sl += v * la[d];
    sr += v * ra[d];
  }
  el[i] = sl;
  er[i] = sr;
}

// e = w * leaky_relu(el[src]+er[dst]); store + atomic segment-max into m[dst].
__global__ void edge_logits(const float* __restrict__ el,
                            const float* __restrict__ er,
                            const int* __restrict__ src,
                            const int* __restrict__ dst,
                            const float* __restrict__ w,
                            float* __restrict__ ebuf, float* __restrict__ m) {
  int i = blockIdx.x * blockDim.x + threadIdx.x;   // over NEDGES*HEADS
  if (i >= NEDGES * HEADS) return;
  int e = i >> 2, h = i & 3;
  float v = el[src[e] * HEADS + h] + er[dst[e] * HEADS + h];
  v = (v > 0.0f) ? v : v * NEG_SLOPE;
  v *= w[e];
  ebuf[i] = v;
  atomicMaxF32(&m[dst[e] * HEADS + h], v);
}

// a = exp(e - m[dst]); atomic segment-sum into s[dst].
__global__ void edge_exp(const int* __restrict__ dst,
                         float* __restrict__ ebuf,
                         const float* __restrict__ m, float* __restrict__ s) {
  int i = blockIdx.x * blockDim.x + threadIdx.x;
  if (i >= NEDGES * HEADS) return;
  int e = i >> 2, h = i & 3;
  float a = __expf(ebuf[i] - m[dst[e] * HEADS + h]);
  ebuf[i] = a;
  atomicAdd(&s[dst[e] * HEADS + h], a);
}

// rst[dst,h,d] += (a[e,h]/s[dst,h]) * ft[src,h,d]; one thread per (edge, 4 dims).
// ft gathered as b128; hout has row stride 256 (concat buffer half).
__global__ void aggregate4(const int* __restrict__ src,
                           const int* __restrict__ dst,
                           const float* __restrict__ ebuf,
                           const float* __restrict__ sinv,
                           const float* __restrict__ ft,
                           float* __restrict__ hout) {
  int i = blockIdx.x * blockDim.x + threadIdx.x;   // over NEDGES*32
  if (i >= NEDGES * 32) return;
  int e = i >> 5;
  int q = i & 31;          // which float4 of the 128-wide row; head = q/8
  int h = q >> 3;
  int se = src[e], de = dst[e];
  float coef = ebuf[e * HEADS + h] * sinv[de * HEADS + h];
  float4 f = reinterpret_cast<const float4*>(ft + (size_t)se * HD)[q];
  float* o = hout + (size_t)de * 256 + q * 4;
  atomicAdd(o + 0, coef * f.x);
  atomicAdd(o + 1, coef * f.y);
  atomicAdd(o + 2, coef * f.z);
  atomicAdd(o + 3, coef * f.w);
}

// in-place ReLU (b128) on one 128-wide column slice of the [N,256] concat buf
__global__ void relu_slice4(float* __restrict__ h, int colofs) {
  int i = blockIdx.x * blockDim.x + threadIdx.x;   // over NNODES*32
  if (i >= NNODES * 32) return;
  int r = i >> 5, q = i & 31;
  float4* p = reinterpret_cast<float4*>(h + (size_t)r * 256 + colofs) + q;
  float4 v = *p;
  v.x = v.x > 0.0f ? v.x : 0.0f;
  v.y = v.y > 0.0f ? v.y : 0.0f;
  v.z = v.z > 0.0f ? v.z : 0.0f;
  v.w = v.w > 0.0f ? v.w : 0.0f;
  *p = v;
}

// ---------------------------------------------------------------------------
extern "C" void kernel_launch(void* const* d_in, const int* in_sizes, int n_in,
                              void* d_out, int out_size, void* d_ws, size_t ws_size,
                              hipStream_t stream) {
  const float* features = (const float*)d_in[0];   // [N,256]
  const int*   src      = (const int*)  d_in[1];   // [E]
  const int*   dst      = (const int*)  d_in[2];   // [E]
  const float* w        = (const float*)d_in[3];   // [E]
  const float* W1       = (const float*)d_in[4];   // [256,128]
  const float* al1      = (const float*)d_in[5];   // [4,32]
  const float* ar1      = (const float*)d_in[6];
  const float* W2       = (const float*)d_in[7];   // [128,128]
  const float* al2      = (const float*)d_in[8];
  const float* ar2      = (const float*)d_in[9];
  const float* Wout     = (const float*)d_in[10];  // [256,128]
  const float* bout     = (const float*)d_in[11];  // [128]
  float* out = (float*)d_out;                      // [N,128]

  // workspace layout (floats)
  float* ft   = (float*)d_ws;                          // N*128
  float* el   = ft   + (size_t)NNODES * HD;            // N*H
  float* er   = el   + (size_t)NNODES * HEADS;         // N*H
  float* mbuf = er   + (size_t)NNODES * HEADS;         // N*H
  float* sbuf = mbuf + (size_t)NNODES * HEADS;         // N*H
  float* ebuf = sbuf + (size_t)NNODES * HEADS;         // E*H
  float* hbuf = ebuf + (size_t)NEDGES * HEADS;         // N*256 (concat x1|x2)

  const int nh   = NNODES * HEADS;      // 200,000
  const int eh   = NEDGES * HEADS;      // 1,600,000
  const int nq   = NNODES * 32;         // 1,600,000  (N*128/4)
  const int eq   = NEDGES * 32;         // 12,800,000 (E*128/4)

  // zero concat buffer once; both layers scatter-add into their half
  {
    int n4 = NNODES * 256 / 4;
    fill4_f32<<<(n4 + 255) / 256, 256, 0, stream>>>((float4*)hbuf, 0.0f, n4);
  }

  for (int layer = 0; layer < 2; ++layer) {
    const float* al  = (layer == 0) ? al1 : al2;
    const float* ar  = (layer == 0) ? ar1 : ar2;
    const int colofs = (layer == 0) ? 0 : 128;

    // ft = A @ W   (fp32 WMMA; lda=256 both layers: features row = 256,
    // layer-2 reads x1 = left half of the [N,256] concat buffer)
    if (layer == 0)
      gemm_wmma_f32<256><<<dim3(HD / 16, NNODES / 16), 32, 0, stream>>>(
          features, W1, ft, nullptr);
    else
      gemm_wmma_f32<128><<<dim3(HD / 16, NNODES / 16), 32, 0, stream>>>(
          hbuf, W2, ft, nullptr);

    attn_logits<<<(nh + 255) / 256, 256, 0, stream>>>(ft, al, ar, el, er);
    init_ms<<<(nh + 255) / 256, 256, 0, stream>>>(mbuf, sbuf, nh);

    edge_logits<<<(eh + 255) / 256, 256, 0, stream>>>(el, er, src, dst, w, ebuf, mbuf);
    edge_exp   <<<(eh + 255) / 256, 256, 0, stream>>>(dst, ebuf, mbuf, sbuf);
    recip_f32  <<<(nh + 255) / 256, 256, 0, stream>>>(sbuf, nh);

    aggregate4<<<(eq + 255) / 256, 256, 0, stream>>>(src, dst, ebuf, sbuf, ft,
                                                     hbuf + colofs);
    relu_slice4<<<(nq + 255) / 256, 256, 0, stream>>>(hbuf, colofs);
  }

  // out = concat(x1,x2) @ Wout + bout   (A = hbuf, lda=256, K=256)
  gemm_wmma_f32<256><<<dim3(128 / 16, NNODES / 16), 32, 0, stream>>>(
      hbuf, Wout, out, bout);

  (void)in_sizes; (void)n_in; (void)out_size; (void)ws_size;
}